// TensorNetInteraction_51110110822523
// MI455X (gfx1250) — compile-verified
//
#include <hip/hip_runtime.h>

// ---------------------------------------------------------------------------
// TensorNet interaction, MI455X (gfx1250, wave32).
// Edge MLP via V_WMMA_F32_16X16X4_F32; scatter collapsed to per-node ea-sums
// (gather index == scatter index); node math fully fused.
// Round-3 changes: weights staged in padded LDS (kills the scratch spills the
// round-2 register cap induced; B fragments now conflict-free ds_load_b64),
// grid-stride edge loop amortizes staging, silu/norm use v_rcp_f32 instead of
// the IEEE divide expansion.
// ---------------------------------------------------------------------------

typedef float v2f __attribute__((ext_vector_type(2)));
typedef float v8f __attribute__((ext_vector_type(8)));

__device__ __forceinline__ v8f wmma_f32_4(v2f a, v2f b, v8f c) {
  // 8-arg form: (neg_a, A, neg_b, B, c_mod, C, reuse_a, reuse_b)
  return __builtin_amdgcn_wmma_f32_16x16x4_f32(false, a, false, b, (short)0, c,
                                               false, false);
}

__device__ __forceinline__ v8f splat8(float x) {
  v8f v = {x, x, x, x, x, x, x, x};
  return v;
}

__device__ __forceinline__ float fast_rcp(float x) {
  return __builtin_amdgcn_rcpf(x);  // v_rcp_f32, ~1 ulp
}

__device__ __forceinline__ float silu(float x) {
  return x * fast_rcp(1.0f + __expf(-x));
}

// D(16x16 tile) += A(16xK, LDS, row stride sA) * W^T (W [nout x K], stride sW;
// LDS or global). ncol = output column owned by this lane.
template <int K>
__device__ __forceinline__ v8f gemm_tile(const float* A, int sA,
                                         const float* W, int sW, int ncol,
                                         int m, int half, v8f acc) {
#pragma unroll
  for (int kk = 0; kk < K; kk += 4) {
    v2f a = *(const v2f*)(A + m * sA + kk + 2 * half);  // even strides: 8B ok
    v2f b = *(const v2f*)(W + ncol * sW + kk + 2 * half);
    acc = wmma_f32_4(a, b, acc);
  }
  return acc;
}

// ---------------------------------------------------------------------------
__global__ __launch_bounds__(256) void zero_f32(float* __restrict__ p, int n) {
  int i = blockIdx.x * 256 + threadIdx.x;
  if (i < n) p[i] = 0.0f;
}

// ---------------------------------------------------------------------------
// Edge kernel: 2 waves/block, 16 edges/wave, grid-stride over edge tiles.
//   silu(silu(silu(attr@W1^T+b1)@W2^T+b2)@W3^T+b3) * cutoff  -> atomic add
//   into sumEA[dst*96 + j]   (j = unit*3 + component)
// ---------------------------------------------------------------------------
#define EW 2
__global__ __launch_bounds__(EW * 32, 4) void edge_kernel(
    const float* __restrict__ eattr, const int* __restrict__ eidx,
    const float* __restrict__ ew, const float* __restrict__ W1,
    const float* __restrict__ b1, const float* __restrict__ W2,
    const float* __restrict__ b2, const float* __restrict__ W3,
    const float* __restrict__ b3, float* __restrict__ sumEA, int E) {
  __shared__ __align__(16) float sW1[32 * 34];   // (32,32) padded
  __shared__ __align__(16) float sW2[64 * 34];   // (64,32) padded
  __shared__ __align__(16) float sW3[96 * 66];   // (96,64) padded
  __shared__ float sBias[192];                   // b1 | b2 | b3
  __shared__ __align__(16) float sA[EW][16 * 66];  // x (32) / h2 (64)
  __shared__ __align__(16) float sB[EW][16 * 98];  // h1 (32) / ea (96)
  __shared__ float sCC[EW][16];
  __shared__ int sDST[EW][16];

  const int tid = threadIdx.x, wave = tid >> 5, lane = tid & 31;
  const int m = lane & 15, half = lane >> 4;
  float* A = sA[wave];
  float* B = sB[wave];

  // stage weights once per block (padded strides -> conflict-free b64 reads)
  for (int i = tid; i < 32 * 32; i += EW * 32)
    sW1[(i >> 5) * 34 + (i & 31)] = W1[i];
  for (int i = tid; i < 64 * 32; i += EW * 32)
    sW2[(i >> 5) * 34 + (i & 31)] = W2[i];
  for (int i = tid; i < 96 * 64; i += EW * 32)
    sW3[(i >> 6) * 66 + (i & 63)] = W3[i];
  for (int i = tid; i < 192; i += EW * 32)
    sBias[i] = (i < 32) ? b1[i] : (i < 96) ? b2[i - 32] : b3[i - 96];
  __syncthreads();

  for (long long base = (long long)blockIdx.x * (EW * 16); base < E;
       base += (long long)gridDim.x * (EW * 16)) {
    const int e0 = (int)base + wave * 16;

    // coalesced 16x32 attr tile load
#pragma unroll
    for (int r = 0; r < 16; ++r) {
      int e = e0 + r;
      A[r * 66 + lane] = (e < E) ? eattr[(long long)e * 32 + lane] : 0.0f;
    }
    if (lane < 16) {
      int e = e0 + lane;
      float w = (e < E) ? ew[e] : 1.0e9f;
      sCC[wave][lane] =
          (w < 5.0f) ? 0.5f * (__cosf(w * 0.6283185307f) + 1.0f) : 0.0f;
      sDST[wave][lane] = (e < E) ? eidx[E + e] : 0;  // row 1 of (2,E)
    }
    __syncthreads();

    // layer 1: (16x32)@(32->32), out -> B[:,0:32]
#pragma unroll
    for (int nt = 0; nt < 2; ++nt) {
      int ncol = nt * 16 + m;
      v8f acc = splat8(sBias[ncol]);
      acc = gemm_tile<32>(A, 66, sW1, 34, ncol, m, half, acc);
#pragma unroll
      for (int i = 0; i < 8; ++i) B[(i + 8 * half) * 98 + ncol] = silu(acc[i]);
    }
    __syncthreads();

    // layer 2: (16x32)@(32->64), out -> A[:,0:64]
#pragma unroll
    for (int nt = 0; nt < 4; ++nt) {
      int ncol = nt * 16 + m;
      v8f acc = splat8(sBias[32 + ncol]);
      acc = gemm_tile<32>(B, 98, sW2, 34, ncol, m, half, acc);
#pragma unroll
      for (int i = 0; i < 8; ++i) A[(i + 8 * half) * 66 + ncol] = silu(acc[i]);
    }
    __syncthreads();

    // layer 3: (16x64)@(64->96), out -> B[:,0:96]
#pragma unroll
    for (int nt = 0; nt < 6; ++nt) {
      int ncol = nt * 16 + m;
      v8f acc = splat8(sBias[96 + ncol]);
      acc = gemm_tile<64>(A, 66, sW3, 66, ncol, m, half, acc);
#pragma unroll
      for (int i = 0; i < 8; ++i) B[(i + 8 * half) * 98 + ncol] = silu(acc[i]);
    }
    __syncthreads();

    // scatter: 16 edges x 96 values -> sumEA (L2-resident, f32 atomics)
#pragma unroll
    for (int it = 0; it < 48; ++it) {
      int flat = it * 32 + lane;  // 0..1535
      int r = flat / 96, c = flat % 96;
      int e = e0 + r;
      if (e < E) {
        float v = B[r * 98 + c] * sCC[wave][r];
        unsafeAtomicAdd(&sumEA[(long long)sDST[wave][r] * 96 + c], v);
      }
    }
    __syncthreads();
  }
}

// ---------------------------------------------------------------------------
// Node kernel: 16 nodes/block, 256 threads (8 waves). Fully fused:
// normalize -> decompose (t, a01,a02,a12, s00,s01,s02,s11,s12) -> 9 WMMA
// GEMMs (Wt0..2) -> Y, msg = I'*sI + A'*sA + S'*sS -> M = Y@msg+msg@Y ->
// decompose/norm -> 9 WMMA GEMMs (Wt3..5) -> dX -> out = Xn + dX + dX@dX.
// ---------------------------------------------------------------------------
__global__ __launch_bounds__(256, 4) void node_kernel(
    const float* __restrict__ X, const float* __restrict__ Wt,
    const float* __restrict__ sumEA, float* __restrict__ out, int N) {
  __shared__ __align__(16) float sIn[9][16 * 34];
  __shared__ __align__(16) float sOut[9][16 * 34];
  const int tid = threadIdx.x, wave = tid >> 5, lane = tid & 31;
  const int m = lane & 15, half = lane >> 4;
  const int n0 = blockIdx.x * 16;
  float xn[2][9];

  // ---- phase 1: normalize + decompose into LDS (rows = nodes, cols = units)
#pragma unroll
  for (int p = 0; p < 2; ++p) {
    int idx = tid + p * 256;
    int ln = idx >> 5, u = idx & 31;
    int n = n0 + ln;
    int nc = (n < N) ? n : (N - 1);
    float x[9];
    float nrm = 1.0f;
#pragma unroll
    for (int q = 0; q < 9; ++q) {
      x[q] = X[((long long)nc * 9 + q) * 32 + u];
      nrm += x[q] * x[q];
    }
    float inv = fast_rcp(nrm);
#pragma unroll
    for (int q = 0; q < 9; ++q) {
      x[q] *= inv;
      xn[p][q] = x[q];
    }
    float t = (x[0] + x[4] + x[8]) * (1.0f / 3.0f);
    int o = ln * 34 + u;
    sIn[0][o] = t;
    sIn[1][o] = 0.5f * (x[1] - x[3]);
    sIn[2][o] = 0.5f * (x[2] - x[6]);
    sIn[3][o] = 0.5f * (x[5] - x[7]);
    sIn[4][o] = x[0] - t;
    sIn[5][o] = 0.5f * (x[1] + x[3]);
    sIn[6][o] = 0.5f * (x[2] + x[6]);
    sIn[7][o] = x[4] - t;
    sIn[8][o] = 0.5f * (x[5] + x[7]);
  }
  __syncthreads();

  // ---- round-1 GEMMs: g=0 -> Wt0 (trace), g=1..3 -> Wt1 (skew), 4..8 -> Wt2
  for (int task = wave; task < 18; task += 8) {
    int g = task >> 1, nt = task & 1;
    const float* W = Wt + (long long)((g > 0) + (g > 3)) * 1024;
    int ncol = nt * 16 + m;
    v8f acc = gemm_tile<32>(&sIn[g][0], 34, W, 32, ncol, m, half, splat8(0.0f));
#pragma unroll
    for (int i = 0; i < 8; ++i) sOut[g][(i + 8 * half) * 34 + ncol] = acc[i];
  }
  __syncthreads();

  // ---- phase 2: build Y & msg, M = Y@msg + msg@Y, renormalized decompose
#pragma unroll
  for (int p = 0; p < 2; ++p) {
    int idx = tid + p * 256;
    int ln = idx >> 5, u = idx & 31;
    int n = n0 + ln;
    int nc = (n < N) ? n : (N - 1);
    int o = ln * 34 + u;
    float Iv = sOut[0][o];
    float a01 = sOut[1][o], a02 = sOut[2][o], a12 = sOut[3][o];
    float s00 = sOut[4][o], s01 = sOut[5][o], s02 = sOut[6][o];
    float s11 = sOut[7][o], s12 = sOut[8][o];
    float s22 = -(s00 + s11);
    float Y[9] = {Iv + s00,  a01 + s01, a02 + s02, s01 - a01, Iv + s11,
                  a12 + s12, s02 - a02, s12 - a12, Iv + s22};
    float sI = sumEA[(long long)nc * 96 + u * 3 + 0];
    float sAc = sumEA[(long long)nc * 96 + u * 3 + 1];
    float sS = sumEA[(long long)nc * 96 + u * 3 + 2];
    float G[9] = {Iv * sI + s00 * sS,   a01 * sAc + s01 * sS,
                  a02 * sAc + s02 * sS, s01 * sS - a01 * sAc,
                  Iv * sI + s11 * sS,   a12 * sAc + s12 * sS,
                  s02 * sS - a02 * sAc, s12 * sS - a12 * sAc,
                  Iv * sI + s22 * sS};
    float M[9];
#pragma unroll
    for (int i = 0; i < 3; ++i)
#pragma unroll
      for (int j = 0; j < 3; ++j) {
        float acc = 0.0f;
#pragma unroll
        for (int k = 0; k < 3; ++k)
          acc += Y[i * 3 + k] * G[k * 3 + j] + G[i * 3 + k] * Y[k * 3 + j];
        M[i * 3 + j] = acc;
      }
    float t2 = (M[0] + M[4] + M[8]) * (1.0f / 3.0f);
    float np = 1.0f;
#pragma unroll
    for (int q = 0; q < 9; ++q) np += M[q] * M[q];
    float ip = fast_rcp(np);
    sIn[0][o] = t2 * ip;
    sIn[1][o] = 0.5f * (M[1] - M[3]) * ip;
    sIn[2][o] = 0.5f * (M[2] - M[6]) * ip;
    sIn[3][o] = 0.5f * (M[5] - M[7]) * ip;
    sIn[4][o] = (M[0] - t2) * ip;
    sIn[5][o] = 0.5f * (M[1] + M[3]) * ip;
    sIn[6][o] = 0.5f * (M[2] + M[6]) * ip;
    sIn[7][o] = (M[4] - t2) * ip;
    sIn[8][o] = 0.5f * (M[5] + M[7]) * ip;
  }
  __syncthreads();

  // ---- round-2 GEMMs with Wt3..5
  for (int task = wave; task < 18; task += 8) {
    int g = task >> 1, nt = task & 1;
    const float* W = Wt + (long long)(3 + (g > 0) + (g > 3)) * 1024;
    int ncol = nt * 16 + m;
    v8f acc = gemm_tile<32>(&sIn[g][0], 34, W, 32, ncol, m, half, splat8(0.0f));
#pragma unroll
    for (int i = 0; i < 8; ++i) sOut[g][(i + 8 * half) * 34 + ncol] = acc[i];
  }
  __syncthreads();

  // ---- phase 3: dX, out = Xn + dX + dX@dX (coalesced over units)
#pragma unroll
  for (int p = 0; p < 2; ++p) {
    int idx = tid + p * 256;
    int ln = idx >> 5, u = idx & 31;
    int n = n0 + ln;
    int o = ln * 34 + u;
    float Iv = sOut[0][o];
    float a01 = sOut[1][o], a02 = sOut[2][o], a12 = sOut[3][o];
    float s00 = sOut[4][o], s01 = sOut[5][o], s02 = sOut[6][o];
    float s11 = sOut[7][o], s12 = sOut[8][o];
    float s22 = -(s00 + s11);
    float dX[9] = {Iv + s00,  a01 + s01, a02 + s02, s01 - a01, Iv + s11,
                   a12 + s12, s02 - a02, s12 - a12, Iv + s22};
    float dd[9];
#pragma unroll
    for (int i = 0; i < 3; ++i)
#pragma unroll
      for (int j = 0; j < 3; ++j) {
        float acc = 0.0f;
#pragma unroll
        for (int k = 0; k < 3; ++k) acc += dX[i * 3 + k] * dX[k * 3 + j];
        dd[i * 3 + j] = acc;
      }
    if (n < N) {
#pragma unroll
      for (int q = 0; q < 9; ++q)
        out[((long long)n * 9 + q) * 32 + u] = xn[p][q] + dX[q] + dd[q];
    }
  }
}

// ---------------------------------------------------------------------------
extern "C" void kernel_launch(void* const* d_in, const int* in_sizes, int n_in,
                              void* d_out, int out_size, void* d_ws,
                              size_t ws_size, hipStream_t stream) {
  const float* X = (const float*)d_in[0];
  const int* eidx = (const int*)d_in[1];
  const float* ew = (const float*)d_in[2];
  const float* eattr = (const float*)d_in[3];
  const float* W1 = (const float*)d_in[4];
  const float* b1 = (const float*)d_in[5];
  const float* W2 = (const float*)d_in[6];
  const float* b2 = (const float*)d_in[7];
  const float* W3 = (const float*)d_in[8];
  const float* b3 = (const float*)d_in[9];
  const float* Wt = (const float*)d_in[10];
  float* out = (float*)d_out;

  const int N = in_sizes[0] / 288;  // X = (N,3,3,32)
  const int E = in_sizes[2];        // edge_weight = (E,)

  float* sumEA = (float*)d_ws;  // N*96 floats (19.2 MB @ N=50k) — L2 resident
  const int nz = N * 96;
  zero_f32<<<(nz + 255) / 256, 256, 0, stream>>>(sumEA, nz);

  int eblocks = (E + EW * 16 - 1) / (EW * 16);
  if (eblocks > 4096) eblocks = 4096;  // grid-stride; amortizes weight staging
  edge_kernel<<<eblocks, EW * 32, 0, stream>>>(eattr, eidx, ew, W1, b1, W2, b2,
                                               W3, b3, sumEA, E);

  node_kernel<<<(N + 15) / 16, 256, 0, stream>>>(X, Wt, sumEA, out, N);
}